// SSNClusteringModel_90013924590150
// MI455X (gfx1250) — compile-verified
//
#include <hip/hip_runtime.h>
#include <math.h>

// SSN soft clustering for MI455X (gfx1250), wave32, WMMA f32 16x16x4.
//
// Channel layout (permuted vs reference; permutation-invariant math):
//   c in [0,768): image_emb channel c
//   c == 768    : x_pos = (p % 24) * 3.0
//   c == 769    : y_pos = (p / 24) * 3.0
//   c == 770    : constant 1.0  (so new_c[:,770] == w = sum_p a)
//   c in [771,784): zero padding  (centroid rows zeroed => no effect)

typedef __attribute__((ext_vector_type(2))) float v2f;
typedef __attribute__((ext_vector_type(8))) float v8f;

#define GRID_   24
#define NPTS    576
#define EMB     768
#define NCLU    9
#define KPAD    16
#define CPAD    784        // 49 * 16
#define NITER   10
#define PT_TILES 36        // 576 / 16
#define N_TILES  49        // 784 / 16
#define NTHREADS 512
#define NWAVES   16
#define SSN_EPS  1e-8f

__device__ __forceinline__ v8f wmma_f32(v2f a, v2f b, v8f c) {
  // D = A(16x4 f32) * B(4x16 f32) + C(16x16 f32)
  return __builtin_amdgcn_wmma_f32_16x16x4_f32(false, a, false, b,
                                               (short)0, c, false, false);
}

template <bool FINAL>
__device__ __forceinline__ void assign_pass(const float* __restrict__ embb,
                                            const float* s_cent,
                                            float* s_a,
                                            const float* s_c2,
                                            float* __restrict__ outb) {
  const int lane = threadIdx.x & 31;
  const int wave = threadIdx.x >> 5;
  const int lm = lane & 15;     // A: row M; B: col N (cluster)
  const int lh = lane >> 4;     // K half select

  for (int tt = wave; tt < PT_TILES; tt += NWAVES) {
    const int p0 = tt * 16;
    const int arow = p0 + lm;                       // point index for A rows
    const float* aro = embb + (size_t)arow * EMB;
    v8f acc = {};

    // channels 0..767 straight from global emb (8 b64 loads in flight)
#pragma unroll 8
    for (int s = 0; s < EMB / 4; ++s) {
      const int c = 4 * s + 2 * lh;
      v2f av = *(const v2f*)(aro + c);
      v2f bv = *(const v2f*)(&s_cent[lm * CPAD + c]);
      acc = wmma_f32(av, bv, acc);
    }
    // channels 768..771: (x, y | 1, 0); centroid cols 770/771 are zero
    {
      v2f av;
      av.x = lh ? 1.0f : (float)(arow % GRID_) * 3.0f;
      av.y = lh ? 0.0f : (float)(arow / GRID_) * 3.0f;
      v2f bv = *(const v2f*)(&s_cent[lm * CPAD + EMB + 2 * lh]);
      acc = wmma_f32(av, bv, acc);
    }

    // scores s = 2*pc - c2 ; C layout: VGPR j -> point p0+j+8*lh, col k = lm
    const float c2k = s_c2[lm];
#pragma unroll
    for (int j = 0; j < 8; ++j) {
      const int pt = p0 + j + 8 * lh;
      s_a[pt * KPAD + lm] = 2.0f * acc[j] - c2k;
    }
    asm volatile("s_wait_dscnt 0x0" ::: "memory");   // same-wave LDS RAW

    if (lane < 16) {                                  // one point per lane
      const int pt = p0 + lane;
      float sv[NCLU];
      float mx = -1e30f;
#pragma unroll
      for (int k = 0; k < NCLU; ++k) {
        sv[k] = s_a[pt * KPAD + k];
        mx = fmaxf(mx, sv[k]);
      }
      float sum = 0.0f;
#pragma unroll
      for (int k = 0; k < NCLU; ++k) { sv[k] = __expf(sv[k] - mx); sum += sv[k]; }
      const float inv = 1.0f / sum;
      if (FINAL) {
#pragma unroll
        for (int k = 0; k < NCLU; ++k)
          outb[(size_t)pt * NCLU + k] = sv[k] * inv;
      } else {
#pragma unroll
        for (int k = 0; k < KPAD; ++k)
          s_a[pt * KPAD + k] = (k < NCLU) ? sv[k] * inv : 0.0f;
      }
    }
  }
}

extern "C" __global__ void __launch_bounds__(NTHREADS, 1)
ssn_cluster_kernel(const float* __restrict__ emb, float* __restrict__ out) {
  const int tid = threadIdx.x;
  const int b = blockIdx.x;
  const float* embb = emb + (size_t)b * NPTS * EMB;
  float* outb = out + (size_t)b * NPTS * NCLU;

  __shared__ float s_cent[KPAD * CPAD];   // centroids (rows 9..15 zero)
  __shared__ float s_newc[KPAD * CPAD];   // new_c accumulator (col 770 = w)
  __shared__ float s_a[NPTS * KPAD];      // soft assignments
  __shared__ float s_c2[KPAD];
  __shared__ float s_red[NTHREADS];

  const int lane = tid & 31;
  const int wave = tid >> 5;
  const int lm = lane & 15;
  const int lh = lane >> 4;

  // ---- init centroids: 8x8 average-pool seeds (k = gy*3 + gx) ----
  for (int idx = tid; idx < KPAD * CPAD; idx += NTHREADS) {
    const int k = idx / CPAD, c = idx - k * CPAD;
    float v = 0.0f;
    if (k < NCLU) {
      if (c < EMB) {
        const int by = (k / 3) * 8, bx = (k % 3) * 8;
        float s = 0.0f;
        for (int dy = 0; dy < 8; ++dy)
          for (int dx = 0; dx < 8; ++dx)
            s += embb[(size_t)((by + dy) * GRID_ + bx + dx) * EMB + c];
        v = s * (1.0f / 64.0f);
      } else if (c == EMB) {
        v = 3.0f * ((float)((k % 3) * 8) + 3.5f);   // mean x over 8x8 block
      } else if (c == EMB + 1) {
        v = 3.0f * ((float)((k / 3) * 8) + 3.5f);   // mean y over 8x8 block
      }
    }
    s_cent[idx] = v;
  }
  __syncthreads();

  for (int it = 0; it <= NITER; ++it) {
    // ---- c2[k] = sum_c centroids[k][c]^2  (channels 0..769) ----
    {
      const int k = tid >> 5, cp = tid & 31;    // 32 threads per cluster row
      float p = 0.0f;
      for (int c = cp; c < EMB + 2; c += 32) {
        const float v = s_cent[k * CPAD + c];
        p += v * v;
      }
      s_red[tid] = p;
      __syncthreads();
      if (tid < KPAD) {
        float s = 0.0f;
#pragma unroll
        for (int i = 0; i < 32; ++i) s += s_red[tid * 32 + i];
        s_c2[tid] = s;
      }
      __syncthreads();
    }

    if (it == NITER) {
      assign_pass<true>(embb, s_cent, s_a, s_c2, outb);
      return;
    }
    assign_pass<false>(embb, s_cent, s_a, s_c2, outb);
    __syncthreads();

    // ---- update: new_c = a^T @ feats ; waves own disjoint channel tiles ----
    for (int nt = wave; nt < N_TILES; nt += NWAVES) {
      const int c0 = nt * 16;
      v8f acc = {};
      if (nt < 48) {
        const float* bcol = embb + (c0 + lm);
#pragma unroll 4
        for (int p0 = 0; p0 < NPTS; p0 += 4) {
          const int pr = p0 + 2 * lh;
          v2f av;
          av.x = s_a[pr * KPAD + lm];
          av.y = s_a[(pr + 1) * KPAD + lm];
          v2f bv;
          bv.x = bcol[(size_t)pr * EMB];
          bv.y = bcol[(size_t)(pr + 1) * EMB];
          acc = wmma_f32(av, bv, acc);
        }
      } else {  // synthesized channels 768..783: x, y, 1, zeros
        for (int p0 = 0; p0 < NPTS; p0 += 4) {
          const int pr = p0 + 2 * lh;
          v2f av;
          av.x = s_a[pr * KPAD + lm];
          av.y = s_a[(pr + 1) * KPAD + lm];
          v2f bv;
          bv.x = (lm == 0) ? (float)(pr % GRID_) * 3.0f
               : (lm == 1) ? (float)(pr / GRID_) * 3.0f
               : (lm == 2) ? 1.0f : 0.0f;
          const int pr1 = pr + 1;
          bv.y = (lm == 0) ? (float)(pr1 % GRID_) * 3.0f
               : (lm == 1) ? (float)(pr1 / GRID_) * 3.0f
               : (lm == 2) ? 1.0f : 0.0f;
          acc = wmma_f32(av, bv, acc);
        }
      }
#pragma unroll
      for (int j = 0; j < 8; ++j)
        s_newc[(j + 8 * lh) * CPAD + c0 + lm] = acc[j];
    }
    __syncthreads();

    // ---- centroids = new_c / (w + eps); zero cols >= 770 and rows >= 9 ----
    for (int idx = tid; idx < KPAD * CPAD; idx += NTHREADS) {
      const int k = idx / CPAD, c = idx - k * CPAD;
      float v = 0.0f;
      if (k < NCLU && c < EMB + 2)
        v = s_newc[idx] / (s_newc[k * CPAD + EMB + 2] + SSN_EPS);
      s_cent[idx] = v;
    }
    __syncthreads();
  }
}

extern "C" void kernel_launch(void* const* d_in, const int* in_sizes, int n_in,
                              void* d_out, int out_size, void* d_ws, size_t ws_size,
                              hipStream_t stream) {
  const float* emb = (const float*)d_in[0];
  float* out = (float*)d_out;
  const int B = in_sizes[0] / (NPTS * EMB);   // 128
  hipLaunchKernelGGL(ssn_cluster_kernel, dim3(B), dim3(NTHREADS), 0, stream,
                     emb, out);
}